// Feature_Selection_Node_34832184770665
// MI455X (gfx1250) — compile-verified
//
#include <hip/hip_runtime.h>
#include <hip/hip_bf16.h>

// Shapes (hardcoded per reference)
#define T_TREES 128
#define B_BATCH 1024
#define D_FEAT  784
#define K_TOP   200

typedef __attribute__((ext_vector_type(2))) float v2f;
typedef __attribute__((ext_vector_type(4))) float v4f;
typedef __attribute__((ext_vector_type(8))) float v8f;

// ---------------------------------------------------------------------------
// Kernel 1: per-tree top-K mask.  att[t,d] = sigmoid(mask[t,d]) if the
// element's stable rank (desc value, ties -> lower index first, matching
// jax.lax.top_k) is < K, else 0.  One block per tree; row cached in LDS.
// ---------------------------------------------------------------------------
__global__ void fsn_topk_attention(const float* __restrict__ mask,
                                   float* __restrict__ att) {
    const int t = blockIdx.x;
    __shared__ float row[D_FEAT];
    for (int d = threadIdx.x; d < D_FEAT; d += blockDim.x)
        row[d] = mask[(size_t)t * D_FEAT + d];
    __syncthreads();

    for (int d = threadIdx.x; d < D_FEAT; d += blockDim.x) {
        const float v = row[d];
        int rank = 0;
        // Uniform inner loop: all lanes read the same LDS word (broadcast).
        for (int j = 0; j < D_FEAT; ++j) {
            const float u = row[j];
            rank += (u > v) || (u == v && j < d);
        }
        const float s = 1.0f / (1.0f + __expf(-v));
        att[(size_t)t * D_FEAT + d] = (rank < K_TOP) ? s : 0.0f;
    }
}

// ---------------------------------------------------------------------------
// Kernel 2: out[b,t,d] = x[b,d] * att[t,d] via WMMA diagonal-matmul.
// Block: 256 threads (8 waves), one b, 16 t, up to 128 d (8 d-tiles).
// Wave w computes C[d0..d0+15, t0..t0+15] = diag(x[b,d0..]) * attT using
// 4x V_WMMA_F32_16X16X4_F32, parks the tile in LDS (padded, conflict-free),
// then the whole block streams 512B-contiguous non-temporal b128 stores.
// ---------------------------------------------------------------------------
#define ROW_PAD 132   // 128 + 4 floats: makes ds_store_b128 conflict-free

__global__ void fsn_expand_wmma(const float* __restrict__ x,
                                const float* __restrict__ att,
                                float* __restrict__ out) {
    __shared__ alignas(16) float tile[16 * ROW_PAD];   // [t_local][d_local]

    const int tid    = threadIdx.x;
    const int lane   = tid & 31;
    const int wave   = tid >> 5;                 // 0..7
    const int m      = lane & 15;                // row index within half-wave
    const int half   = (lane >> 4) & 1;          // 0: K={0,1}, 1: K={2,3}
    const int b      = blockIdx.z;
    const int t0     = blockIdx.y * 16;
    const int dtile  = blockIdx.x * 8 + wave;    // 0..55, valid < 49
    const int dbase  = blockIdx.x * 128;

    if (dtile < (D_FEAT / 16)) {                 // wave-uniform branch
        const int d0 = dtile * 16;

        // x value this lane contributes on the diagonal: x[b, d0+m]
        const float xv = x[(size_t)b * D_FEAT + d0 + m];
        const int   tt = t0 + m;                 // B-matrix column owner

        v8f acc = {};
        #pragma unroll
        for (int c = 0; c < 4; ++c) {
            // A (16x4) chunk c of diag(x): lanes 0-15 hold K={0,1},
            // lanes 16-31 hold K={2,3}.  Nonzero iff m == 4c + K.
            const int k_even = 4 * c + 2 * half;       // K=0 or K=2 slot
            v2f a;
            a.x = (m == k_even)     ? xv : 0.0f;
            a.y = (m == k_even + 1) ? xv : 0.0f;

            // B (4x16): B[k, t] = att[t0+t, d0+4c+k]; same half split.
            const size_t col = (size_t)tt * D_FEAT + d0 + k_even;
            v2f bb;
            bb.x = att[col];
            bb.y = att[col + 1];

            acc = __builtin_amdgcn_wmma_f32_16x16x4_f32(
                false, a, false, bb, (short)0, acc, false, false);
        }

        // Park tile in LDS: lane owns t_local = m, d_local = wave*16+8*half+v
        const int base = m * ROW_PAD + wave * 16 + 8 * half;
        v4f lo = { acc[0], acc[1], acc[2], acc[3] };
        v4f hi = { acc[4], acc[5], acc[6], acc[7] };
        *(v4f*)&tile[base]     = lo;   // ds_store_b128
        *(v4f*)&tile[base + 4] = hi;   // ds_store_b128
    }
    __syncthreads();

    // Epilogue: fully-coalesced non-temporal b128 stores.
    const int validD = min(128, D_FEAT - dbase);     // 128 or 16 (last block)
    #pragma unroll
    for (int s = 0; s < 2; ++s) {
        const int j    = tid + s * 256;              // 0..511 float4 slots
        const int row  = j >> 5;                     // t_local 0..15
        const int col  = (j & 31) * 4;               // d_local
        if (col < validD) {
            v4f v = *(const v4f*)&tile[row * ROW_PAD + col];       // ds_load_b128
            float* dst = out + ((size_t)b * T_TREES + (t0 + row)) * D_FEAT
                             + dbase + col;
            __builtin_nontemporal_store(v, (v4f*)dst);             // b128 NT
        }
    }
}

// ---------------------------------------------------------------------------
extern "C" void kernel_launch(void* const* d_in, const int* in_sizes, int n_in,
                              void* d_out, int out_size, void* d_ws, size_t ws_size,
                              hipStream_t stream) {
    (void)in_sizes; (void)n_in; (void)out_size; (void)d_ws; (void)ws_size;

    const float* x    = (const float*)d_in[0];          // [B, D]
    const float* mask = (const float*)d_in[1];          // [T, D]
    float* out = (float*)d_out;                         // [B,T,D] ++ [T,D]
    float* att = out + (size_t)B_BATCH * T_TREES * D_FEAT;

    // Phase 1: sparse attention mask (also the 2nd tuple output).
    fsn_topk_attention<<<T_TREES, 256, 0, stream>>>(mask, att);

    // Phase 2: 411MB broadcast-multiply expansion (store-bandwidth bound).
    dim3 grid((D_FEAT / 16 + 7) / 8, T_TREES / 16, B_BATCH);  // (7, 8, 1024)
    fsn_expand_wmma<<<grid, 256, 0, stream>>>(x, att, out);
}